// MMConvex_70342974373992
// MI455X (gfx1250) — compile-verified
//
#include <hip/hip_runtime.h>
#include <math.h>

typedef __attribute__((ext_vector_type(2))) float v2f;
typedef __attribute__((ext_vector_type(8))) float v8f;

#define DEVI __device__ __forceinline__

DEVI int clampi(int v, int lo, int hi) { return v < lo ? lo : (v > hi ? hi : v); }

// ---------------------------------------------------------------------------
// stride-2 avg pool, k=3, pad=1, count_include_pad=False (pyramid)
// ---------------------------------------------------------------------------
__global__ void k_pool2(const float* __restrict__ in, float* __restrict__ out,
                        int C, int Di, int Do) {
  int tot = C * Do * Do * Do;
  int idx = blockIdx.x * blockDim.x + threadIdx.x;
  if (idx >= tot) return;
  int xo = idx % Do; int t = idx / Do;
  int yo = t % Do; t /= Do;
  int zo = t % Do; int c = t / Do;
  const float* ic = in + (size_t)c * Di * Di * Di;
  float s = 0.f; int cnt = 0;
  for (int dz = -1; dz <= 1; ++dz) { int zi = 2*zo + dz; if (zi < 0 || zi >= Di) continue;
    for (int dy = -1; dy <= 1; ++dy) { int yi = 2*yo + dy; if (yi < 0 || yi >= Di) continue;
      for (int dx = -1; dx <= 1; ++dx) { int xi = 2*xo + dx; if (xi < 0 || xi >= Di) continue;
        s += ic[((size_t)zi * Di + yi) * Di + xi]; ++cnt; } } }
  out[idx] = s / (float)cnt;
}

// ---------------------------------------------------------------------------
// stride-1 avg pool, k=3, pad=1, count_include_pad=True (smoothing)
// ---------------------------------------------------------------------------
__global__ void k_pool1(const float* __restrict__ in, float* __restrict__ out,
                        int C, int n) {
  int V = n * n * n;
  int tot = C * V;
  int idx = blockIdx.x * blockDim.x + threadIdx.x;
  if (idx >= tot) return;
  int v = idx % V; int c = idx / V;
  int x = v % n, y = (v / n) % n, z = v / (n * n);
  const float* ic = in + (size_t)c * V;
  float s = 0.f;
  for (int dz = -1; dz <= 1; ++dz) { int zi = z + dz; if (zi < 0 || zi >= n) continue;
    for (int dy = -1; dy <= 1; ++dy) { int yi = y + dy; if (yi < 0 || yi >= n) continue;
      for (int dx = -1; dx <= 1; ++dx) { int xi = x + dx; if (xi < 0 || xi >= n) continue;
        s += ic[((size_t)zi * n + yi) * n + xi]; } } }
  out[idx] = s * (1.f / 27.f);
}

// ---------------------------------------------------------------------------
// trilinear resize, align_corners=True, cube -> cube
// ---------------------------------------------------------------------------
__global__ void k_resize(const float* __restrict__ in, float* __restrict__ out,
                         int C, int ni, int no_) {
  int Vo = no_ * no_ * no_;
  int tot = C * Vo;
  int idx = blockIdx.x * blockDim.x + threadIdx.x;
  if (idx >= tot) return;
  int v = idx % Vo; int c = idx / Vo;
  int x = v % no_, y = (v / no_) % no_, z = v / (no_ * no_);
  float f = (no_ > 1) ? (float)(ni - 1) / (float)(no_ - 1) : 0.f;
  float pz = z * f, py = y * f, px = x * f;
  int z0 = clampi((int)floorf(pz), 0, ni - 2);
  int y0 = clampi((int)floorf(py), 0, ni - 2);
  int x0 = clampi((int)floorf(px), 0, ni - 2);
  float wz = pz - (float)z0, wy = py - (float)y0, wx = px - (float)x0;
  const float* ic = in + (size_t)c * ni * ni * ni;
  float acc = 0.f;
#pragma unroll
  for (int t = 0; t < 8; ++t) {
    int dz = t >> 2, dy = (t >> 1) & 1, dx = t & 1;
    float w = (dz ? wz : 1.f - wz) * (dy ? wy : 1.f - wy) * (dx ? wx : 1.f - wx);
    acc += w * ic[((size_t)(z0 + dz) * ni + (y0 + dy)) * ni + (x0 + dx)];
  }
  out[idx] = acc;
}

// ---------------------------------------------------------------------------
// 3x3x3 stride-2 pad-1 conv + ReLU as implicit GEMM on V_WMMA_F32_16X16X4_F32
// A = weights (M=16 out-ch x K=Ci*27), B = im2col patch (K x N=16 voxels).
// Both A and B loads are branchless (clamped address + select) so the WMMA
// K-loop is straight-line: clause'd loads + cndmask + wmma.
// ---------------------------------------------------------------------------
__global__ void __launch_bounds__(256)
k_conv3s2_wmma(const float* __restrict__ in, const float* __restrict__ wt,
               float* __restrict__ out, int Ci, int Co, int Di, int Do) {
  int lane = threadIdx.x & 31;
  int wave = threadIdx.x >> 5;
  int TW = (Do + 15) >> 4;
  int tiles = Do * Do * TW;
  int tile = blockIdx.x * (blockDim.x >> 5) + wave;
  if (tile >= tiles) return;              // wave-uniform
  int xt = tile % TW; int tt = tile / TW;
  int yo = tt % Do; int zo = tt / Do;
  int nloc = lane & 15, hi = lane >> 4;
  int xo = (xt << 4) + nloc;
  int xc = xo < Do ? xo : Do - 1;
  int K = Ci * 27;
  int Ks = (K + 3) >> 2;
  int wco = nloc < Co ? nloc : Co - 1;    // clamped weight row
  bool arow = nloc < Co;
  v8f acc = {};
  for (int kk = 0; kk < Ks; ++kk) {
    int kb = (kk << 2) + hi * 2;
    int kb0 = kb < K ? kb : K - 1;
    int kb1 = kb + 1 < K ? kb + 1 : K - 1;
    v2f a, b;
    float a0 = wt[wco * K + kb0];         // always in-bounds
    float a1 = wt[wco * K + kb1];
    a[0] = (arow && kb < K) ? a0 : 0.f;
    a[1] = (arow && kb + 1 < K) ? a1 : 0.f;
#pragma unroll
    for (int u = 0; u < 2; ++u) {
      int kq = kb + u;
      int kqc = kq < K ? kq : K - 1;
      int ci = kqc / 27; int r = kqc % 27;
      int dz = r / 9, dy = (r / 3) % 3, dx = r % 3;
      int zi = 2 * zo + dz - 1, yi = 2 * yo + dy - 1, xi = 2 * xc + dx - 1;
      bool inb = (kq < K) && (xo < Do) &&
                 ((unsigned)zi < (unsigned)Di) &&
                 ((unsigned)yi < (unsigned)Di) &&
                 ((unsigned)xi < (unsigned)Di);
      size_t idx3 = ((size_t)(ci * Di + clampi(zi, 0, Di - 1)) * Di +
                     clampi(yi, 0, Di - 1)) * Di + clampi(xi, 0, Di - 1);
      float val = in[idx3];               // always in-bounds
      b[u] = inb ? val : 0.f;
    }
    acc = __builtin_amdgcn_wmma_f32_16x16x4_f32(false, a, false, b,
                                                (short)0, acc, false, false);
  }
  if (xo < Do) {
#pragma unroll
    for (int r = 0; r < 8; ++r) {
      int co = hi * 8 + r;
      if (co < Co) {
        float v = acc[r];
        out[((size_t)(co * Do + zo) * Do + yo) * Do + xo] = v > 0.f ? v : 0.f;
      }
    }
  }
}

// ---------------------------------------------------------------------------
// concat fine(8ch) with per-voxel L2-normalized coarse(16ch) -> 24ch
// ---------------------------------------------------------------------------
__global__ void k_concat_norm(const float* __restrict__ fine,
                              const float* __restrict__ coarse,
                              float* __restrict__ out, int V) {
  int v = blockIdx.x * blockDim.x + threadIdx.x;
  if (v >= V) return;
#pragma unroll
  for (int c = 0; c < 8; ++c) out[(size_t)c * V + v] = fine[(size_t)c * V + v];
  float tmp[16]; float s = 0.f;
#pragma unroll
  for (int c = 0; c < 16; ++c) { float t = coarse[(size_t)c * V + v]; tmp[c] = t; s += t * t; }
  float nrm = fmaxf(sqrtf(s), 1e-12f);
#pragma unroll
  for (int c = 0; c < 16; ++c) out[(size_t)(8 + c) * V + v] = tmp[c] / nrm;
}

// ---------------------------------------------------------------------------
// SSD correlation volume (direct, 24-channel stages). fp16 cost storage.
// ---------------------------------------------------------------------------
template <int C>
__global__ void k_ssd(const float* __restrict__ fs, const float* __restrict__ ft,
                      _Float16* __restrict__ cost, int n, int k) {
  int V = n * n * n;
  int v = blockIdx.x * blockDim.x + threadIdx.x;
  if (v >= V) return;
  int x = v % n, y = (v / n) % n, z = v / (n * n);
  float ftv[C];
#pragma unroll
  for (int c = 0; c < C; ++c) ftv[c] = ft[(size_t)c * V + v];
  int R = 2 * k + 1;
  int q = 0;
  for (int oz = 0; oz < R; ++oz) {
    int zs = clampi(z + oz - k, 0, n - 1);
    for (int oy = 0; oy < R; ++oy) {
      int ys = clampi(y + oy - k, 0, n - 1);
      for (int ox = 0; ox < R; ++ox, ++q) {
        int xs = clampi(x + ox - k, 0, n - 1);
        size_t base = ((size_t)zs * n + ys) * n + xs;
        float s = 0.f;
#pragma unroll
        for (int c = 0; c < C; ++c) {
          float d = fs[(size_t)c * V + base] - ftv[c];
          s += d * d;
        }
        cost[(size_t)q * V + v] = (_Float16)s;
      }
    }
  }
}

// ---------------------------------------------------------------------------
// SSD correlation volume, fine_only stage (C=8, k=3), LDS halo tile staged
// with CDNA5 async global->LDS DMA (ASYNCcnt-tracked, no VGPR round trip).
// Tile: 8x8x4 voxels, halo 14x14x10 x 8ch = 62720 B of LDS.
// ---------------------------------------------------------------------------
__global__ void __launch_bounds__(256)
k_ssd_tile(const float* __restrict__ fs, const float* __restrict__ ft,
           _Float16* __restrict__ cost, int n) {
  const int KR = 3, R = 7;
  const int EX = 14, EY = 14, EZ = 10;   // 8+2k, 8+2k, 4+2k
  const int EPC = EZ * EY * EX;          // 1960 per channel
  const int TOT = EPC * 8;               // 15680 floats
  __shared__ float tile[TOT];
  int V = n * n * n;
  int nbx = n >> 3, nby = n >> 3;
  int bid = blockIdx.x;
  int bx = bid % nbx; int by = (bid / nbx) % nby; int bz = bid / (nbx * nby);
  int x0 = bx << 3, y0 = by << 3, z0 = bz << 2;
  int tid = threadIdx.x;
  // ---- stage halo (edge-clamped) global -> LDS via async DMA ----
  for (int i = tid; i < TOT; i += 256) {
    int c = i / EPC; int rem = i % EPC;
    int ez = rem / (EY * EX); int r2 = rem % (EY * EX);
    int ey = r2 / EX; int ex = r2 % EX;
    int gz = clampi(z0 + ez - KR, 0, n - 1);
    int gy = clampi(y0 + ey - KR, 0, n - 1);
    int gx = clampi(x0 + ex - KR, 0, n - 1);
    const float* gp = fs + (size_t)c * V + ((size_t)gz * n + gy) * n + gx;
    unsigned lp = (unsigned)(size_t)&tile[i];   // low 32 bits = LDS address
    asm volatile("global_load_async_to_lds_b32 %0, %1, off"
                 :: "v"(lp), "v"(gp) : "memory");
  }
  asm volatile("s_wait_asynccnt 0x0" ::: "memory");
  __syncthreads();
  // ---- compute: each thread owns one voxel, all 343 offsets ----
  int lx = tid & 7, ly = (tid >> 3) & 7, lz = tid >> 6;
  int x = x0 + lx, y = y0 + ly, z = z0 + lz;
  int v = ((z)*n + y) * n + x;
  float ftv[8];
#pragma unroll
  for (int c = 0; c < 8; ++c) ftv[c] = ft[(size_t)c * V + v];
  int q = 0;
  for (int oz = 0; oz < R; ++oz)
    for (int oy = 0; oy < R; ++oy)
      for (int ox = 0; ox < R; ++ox, ++q) {
        int base = ((lz + oz) * EY + (ly + oy)) * EX + (lx + ox);
        float s = 0.f;
#pragma unroll
        for (int c = 0; c < 8; ++c) {
          float d = tile[c * EPC + base] - ftv[c];
          s += d * d;
        }
        cost[(size_t)q * V + v] = (_Float16)s;
      }
}

// ---------------------------------------------------------------------------
// Fused coupled_convex step: argmin_q( cost[q,v] + coef*||mesh_q - d0_v||^2 )
// Cross term -2*mesh.d0 via V_WMMA_F32_16X16X4_F32 (M=offsets, N=voxels, K=3)
// One wave owns 16 voxels; streams the fp16 cost volume exactly once.
// ---------------------------------------------------------------------------
__global__ void __launch_bounds__(256)
k_argmin(const _Float16* __restrict__ cost, const float* __restrict__ d0,
         float* __restrict__ outdisp, int n, int k, int Q, float coef, int use_d0) {
  int V = n * n * n;
  int lane = threadIdx.x & 31;
  int wave = threadIdx.x >> 5;
  int g = blockIdx.x * (blockDim.x >> 5) + wave;
  int vbase = g << 4;
  if (vbase >= V) return;                 // wave-uniform
  int nloc = lane & 15, hi = lane >> 4;
  int v = vbase + nloc;
  float dz = 0.f, dy = 0.f, dx = 0.f;
  if (use_d0) {
    dz = d0[v];
    dy = d0[(size_t)V + v];
    dx = d0[2 * (size_t)V + v];
  }
  float d0sq = dz * dz + dy * dy + dx * dx;
  v2f b; b[0] = hi ? dx : dz; b[1] = hi ? 0.f : dy;   // B rows k = (z,y,x,0)
  float sc = 2.f / (float)(n - 1);
  int R = 2 * k + 1, R2 = R * R;
  float best = 3.4e38f; int bidx = 0;
  int QT = (Q + 15) >> 4;
  for (int qt = 0; qt < QT; ++qt) {
    int qa = (qt << 4) + nloc;
    int qac = qa < Q ? qa : Q - 1;
    int qz0 = qac / R2, qy0 = (qac / R) % R, qx0 = qac % R;
    float mz0 = (qz0 - k) * sc, my0 = (qy0 - k) * sc, mx0 = (qx0 - k) * sc;
    v2f a;
    a[0] = (qa < Q) ? (-2.f * (hi ? mx0 : mz0)) : 0.f;   // fold -2 into A
    a[1] = (qa < Q && !hi) ? (-2.f * my0) : 0.f;
    v8f cr = {};
    cr = __builtin_amdgcn_wmma_f32_16x16x4_f32(false, a, false, b,
                                               (short)0, cr, false, false);
    if (qt + 1 < QT) {
      int qn = ((qt + 1) << 4) + hi * 8;
      if (qn < Q) __builtin_prefetch(&cost[(size_t)qn * V + v], 0, 1);
    }
#pragma unroll
    for (int r = 0; r < 8; ++r) {
      int q2 = (qt << 4) + hi * 8 + r;
      int q2c = q2 < Q ? q2 : Q - 1;
      int qz = q2c / R2, qy = (q2c / R) % R, qx = q2c % R;
      float mz = (qz - k) * sc, my = (qy - k) * sc, mx = (qx - k) * sc;
      float m2 = mz * mz + my * my + mx * mx;
      float cv = (float)cost[(size_t)q2c * V + v];      // branchless load
      float score = cv + coef * (m2 + cr[r] + d0sq);
      if (q2 < Q && score < best) { best = score; bidx = q2; }
    }
  }
  // merge the two half-wave offset ranges (first-occurrence tie break)
  float ob = __shfl_xor(best, 16, 32);
  int oi = __shfl_xor(bidx, 16, 32);
  if (ob < best || (ob == best && oi < bidx)) { best = ob; bidx = oi; }
  if (hi == 0) {
    int qz = bidx / R2, qy = (bidx / R) % R, qx = bidx % R;
    outdisp[v] = (qz - k) * sc;
    outdisp[(size_t)V + v] = (qy - k) * sc;
    outdisp[2 * (size_t)V + v] = (qx - k) * sc;
  }
}

// ---------------------------------------------------------------------------
// grid_sample with identity grid + displacement; optional border clamp and
// optional "+= disp" (displacement composition). src/out share cube dims.
// Branchless taps: clamped gather index, weight masked by validity.
// ---------------------------------------------------------------------------
template <int C>
__global__ void k_warp(const float* __restrict__ src, const float* __restrict__ disp,
                       float* __restrict__ out, int n, int border, int add_disp) {
  int V = n * n * n;
  int v = blockIdx.x * blockDim.x + threadIdx.x;
  if (v >= V) return;
  int x = v % n, y = (v / n) % n, z = v / (n * n);
  float hf = 0.5f * (float)(n - 1);
  float pz = z + disp[v] * hf;
  float py = y + disp[(size_t)V + v] * hf;
  float px = x + disp[2 * (size_t)V + v] * hf;
  if (border) {
    pz = fminf(fmaxf(pz, 0.f), (float)(n - 1));
    py = fminf(fmaxf(py, 0.f), (float)(n - 1));
    px = fminf(fmaxf(px, 0.f), (float)(n - 1));
  }
  int z0 = (int)floorf(pz), y0 = (int)floorf(py), x0 = (int)floorf(px);
  float wz = pz - (float)z0, wy = py - (float)y0, wx = px - (float)x0;
  float acc[C];
#pragma unroll
  for (int c = 0; c < C; ++c) acc[c] = 0.f;
#pragma unroll
  for (int t = 0; t < 8; ++t) {
    int dzt = t >> 2, dyt = (t >> 1) & 1, dxt = t & 1;
    int zi = z0 + dzt, yi = y0 + dyt, xi = x0 + dxt;
    float w = (dzt ? wz : 1.f - wz) * (dyt ? wy : 1.f - wy) * (dxt ? wx : 1.f - wx);
    bool valid = ((unsigned)zi < (unsigned)n) && ((unsigned)yi < (unsigned)n) &&
                 ((unsigned)xi < (unsigned)n);
    float wv = valid ? w : 0.f;
    size_t idx3 = ((size_t)clampi(zi, 0, n - 1) * n + clampi(yi, 0, n - 1)) * n +
                  clampi(xi, 0, n - 1);
#pragma unroll
    for (int c = 0; c < C; ++c) acc[c] += wv * src[(size_t)c * V + idx3];
  }
#pragma unroll
  for (int c = 0; c < C; ++c)
    out[(size_t)c * V + v] = acc[c] + (add_disp ? disp[(size_t)c * V + v] : 0.f);
}

// ---------------------------------------------------------------------------
// phi = identity grid + ds, written to both output slots
// ---------------------------------------------------------------------------
__global__ void k_phi(const float* __restrict__ ds, float* __restrict__ out1,
                      float* __restrict__ out2, int n) {
  int V = n * n * n;
  int tot = 3 * V;
  int idx = blockIdx.x * blockDim.x + threadIdx.x;
  if (idx >= tot) return;
  int v = idx % V; int c = idx / V;
  int x = v % n, y = (v / n) % n, z = v / (n * n);
  int ic = (c == 0) ? z : ((c == 1) ? y : x);
  float g = -1.f + 2.f * (float)ic / (float)(n - 1);
  float val = g + ds[idx];
  out1[idx] = val;
  out2[idx] = val;
}

// ---------------------------------------------------------------------------
static inline int cdiv(int a, int b) { return (a + b - 1) / b; }

extern "C" void kernel_launch(void* const* d_in, const int* in_sizes, int n_in,
                              void* d_out, int out_size, void* d_ws, size_t ws_size,
                              hipStream_t stream) {
  (void)in_sizes; (void)n_in; (void)out_size; (void)ws_size;
  const float* src = (const float*)d_in[0];
  const float* tgt = (const float*)d_in[1];
  const float* W1 = (const float*)d_in[2];   // (8,1,3,3,3)
  const float* W2 = (const float*)d_in[3];   // (16,8,3,3,3)
  float* out = (float*)d_out;

  const int N0 = 128, N2 = 64, N4 = 32;
  const size_t V0 = (size_t)N0 * N0 * N0;
  const size_t V2 = (size_t)N2 * N2 * N2;
  const size_t V4 = (size_t)N4 * N4 * N4;

  size_t cur = 0;
  auto alloc = [&](size_t bytes) -> void* {
    void* p = (char*)d_ws + cur;
    cur += (bytes + 255) & ~(size_t)255;
    return p;
  };
  float* t2  = (float*)alloc(V2 * 4);
  float* w2  = (float*)alloc(V2 * 4);
  float* t4  = (float*)alloc(V4 * 4);
  float* w4  = (float*)alloc(V4 * 4);
  float* ff_s = (float*)alloc(8 * V2 * 4);     // fine feats (max 8 x 64^3)
  float* ff_t = (float*)alloc(8 * V2 * 4);
  float* cf_s = (float*)alloc(16 * 4096 * 4);  // coarse feats (max 16 x 16^3)
  float* cf_t = (float*)alloc(16 * 4096 * 4);
  float* cr_s = (float*)alloc(16 * V4 * 4);    // coarse resized (max 16 x 32^3)
  float* cr_t = (float*)alloc(16 * V4 * 4);
  float* fsb = (float*)alloc(24 * V4 * 4);     // concat (max 24 x 32^3)
  float* ftb = (float*)alloc(24 * V4 * 4);
  _Float16* cost = (_Float16*)alloc((size_t)343 * V2 * 2);  // fp16 cost volume
  float* draw = (float*)alloc(3 * V2 * 4);
  float* dcc0 = (float*)alloc(3 * V2 * 4);
  float* dcc1 = (float*)alloc(3 * V2 * 4);
  float* d4  = (float*)alloc(3 * V4 * 4);
  float* u4  = (float*)alloc(3 * V2 * 4);
  float* w2w = (float*)alloc(V2 * 4);
  float* dB  = (float*)alloc(3 * V2 * 4);
  float* d2  = (float*)alloc(3 * V2 * 4);
  float* u2  = (float*)alloc(3 * V0 * 4);
  float* warped = (float*)alloc(V0 * 4);
  float* tA = (float*)alloc(3 * V0 * 4);
  float* tB = (float*)alloc(3 * V0 * 4);
  float* dsb = (float*)alloc(3 * V0 * 4);

  auto pool2 = [&](const float* i, float* o, int C, int Di) {
    int Do = (Di - 1) / 2 + 1;
    int tot = C * Do * Do * Do;
    k_pool2<<<cdiv(tot, 256), 256, 0, stream>>>(i, o, C, Di, Do);
  };
  auto pool1 = [&](const float* i, float* o, int C, int n) {
    int tot = C * n * n * n;
    k_pool1<<<cdiv(tot, 256), 256, 0, stream>>>(i, o, C, n);
  };
  auto conv = [&](const float* i, const float* w, float* o, int Ci, int Co, int Di) {
    int Do = (Di - 1) / 2 + 1;
    int TW = cdiv(Do, 16);
    int tiles = Do * Do * TW;
    k_conv3s2_wmma<<<cdiv(tiles, 8), 256, 0, stream>>>(i, w, o, Ci, Co, Di, Do);
  };
  auto resize = [&](const float* i, float* o, int C, int ni, int no_) {
    int tot = C * no_ * no_ * no_;
    k_resize<<<cdiv(tot, 256), 256, 0, stream>>>(i, o, C, ni, no_);
  };
  auto warp1 = [&](const float* s, const float* d, float* o, int n, int border, int add) {
    int tot = n * n * n;
    k_warp<1><<<cdiv(tot, 256), 256, 0, stream>>>(s, d, o, n, border, add);
  };
  auto warp3 = [&](const float* s, const float* d, float* o, int n, int border, int add) {
    int tot = n * n * n;
    k_warp<3><<<cdiv(tot, 256), 256, 0, stream>>>(s, d, o, n, border, add);
  };
  auto ssd24 = [&](const float* f_s, const float* f_t, int n, int k) {
    int tot = n * n * n;
    k_ssd<24><<<cdiv(tot, 256), 256, 0, stream>>>(f_s, f_t, cost, n, k);
  };
  auto coupled = [&](int n, int k) {            // result left in dcc0 (3, n^3)
    int R = 2 * k + 1; int Q = R * R * R;
    int Vn = n * n * n;
    int blocks = cdiv(cdiv(Vn, 16), 8);
    k_argmin<<<blocks, 256, 0, stream>>>(cost, dcc0, draw, n, k, Q, 0.f, 0);
    pool1(draw, dcc0, 3, n);
    const float coefs[6] = {0.003f, 0.01f, 0.03f, 0.1f, 0.3f, 1.0f};
    for (int i = 0; i < 6; ++i) {
      k_argmin<<<blocks, 256, 0, stream>>>(cost, dcc0, draw, n, k, Q, coefs[i], 1);
      pool1(draw, dcc0, 3, n);
      pool1(dcc0, dcc1, 3, n);
      pool1(dcc1, dcc0, 3, n);
    }
  };

  // ---------------- pyramid ----------------
  pool2(src, w2, 1, N0); pool2(tgt, t2, 1, N0);
  pool2(w2, w4, 1, N2);  pool2(t2, t4, 1, N2);

  // ---------------- stage A: _mconvex(w4, t4, k=2) at 32^3 ----------------
  conv(t4, W1, ff_t, 1, 8, 32);          // fine 16^3 x8
  conv(w4, W1, ff_s, 1, 8, 32);
  conv(ff_t, W2, cf_t, 8, 16, 16);       // coarse 8^3 x16
  conv(ff_s, W2, cf_s, 8, 16, 16);
  resize(cf_s, cr_s, 16, 8, 16);
  resize(cf_t, cr_t, 16, 8, 16);
  k_concat_norm<<<cdiv(4096, 256), 256, 0, stream>>>(ff_s, cr_s, fsb, 4096);
  k_concat_norm<<<cdiv(4096, 256), 256, 0, stream>>>(ff_t, cr_t, ftb, 4096);
  ssd24(fsb, ftb, 16, 2);
  coupled(16, 2);
  resize(dcc0, tA, 3, 16, 32);
  pool1(tA, d4, 3, 32); pool1(d4, tB, 3, 32); pool1(tB, d4, 3, 32);   // smooth3
  resize(d4, u4, 3, 32, 64);
  warp1(w2, u4, w2w, 64, 0, 0);

  // ---------------- stage B: _mconvex(w2w, t2, k=3) at 64^3 ----------------
  conv(t2, W1, ff_t, 1, 8, 64);          // fine 32^3 x8
  conv(w2w, W1, ff_s, 1, 8, 64);
  conv(ff_t, W2, cf_t, 8, 16, 32);       // coarse 16^3 x16
  conv(ff_s, W2, cf_s, 8, 16, 32);
  resize(cf_s, cr_s, 16, 16, 32);
  resize(cf_t, cr_t, 16, 16, 32);
  k_concat_norm<<<cdiv((int)V4, 256), 256, 0, stream>>>(ff_s, cr_s, fsb, (int)V4);
  k_concat_norm<<<cdiv((int)V4, 256), 256, 0, stream>>>(ff_t, cr_t, ftb, (int)V4);
  ssd24(fsb, ftb, 32, 3);
  coupled(32, 3);
  resize(dcc0, tA, 3, 32, 64);
  pool1(tA, dB, 3, 64); pool1(dB, tB, 3, 64); pool1(tB, dB, 3, 64);   // smooth3
  warp3(u4, dB, d2, 64, 1, 1);           // compose(u4, dB)
  resize(d2, u2, 3, 64, 128);
  warp1(src, u2, warped, 128, 0, 0);

  // ---------------- stage C: _mconvex(warped, tgt, k=3, fine_only) at 128^3 --
  conv(tgt, W1, ff_t, 1, 8, 128);        // fine 64^3 x8
  conv(warped, W1, ff_s, 1, 8, 128);
  {                                      // LDS-tiled SSD with async->LDS staging
    int blocks = (64 / 8) * (64 / 8) * (64 / 4);
    k_ssd_tile<<<blocks, 256, 0, stream>>>(ff_s, ff_t, cost, 64);
  }
  coupled(64, 3);
  resize(dcc0, tA, 3, 64, 128);
  pool1(tA, tB, 3, 128); pool1(tB, tA, 3, 128); pool1(tA, tB, 3, 128); // smooth3 -> tB
  warp3(u2, tB, tA, 128, 1, 1);          // dh = compose(u2, smooth3(...)) -> tA
  pool1(tA, dsb, 3, 128); pool1(dsb, tB, 3, 128); pool1(tB, dsb, 3, 128); // ds

  // ---------------- outputs: warped, phi, phi ----------------
  warp1(src, dsb, out, 128, 0, 0);                       // grid_sample(src, g+ds)
  k_phi<<<cdiv((int)(3 * V0), 256), 256, 0, stream>>>(dsb, out + V0, out + 4 * V0, 128);
}